// T_RED_GNN_20993800142942
// MI455X (gfx1250) — compile-verified
//
#include <hip/hip_runtime.h>
#include <math.h>

typedef __attribute__((ext_vector_type(2))) float v2f;
typedef __attribute__((ext_vector_type(8))) float v8f;

#define D_EMB 20

__device__ __forceinline__ v8f wmma_f32_4(v2f a, v2f b, v8f c) {
  // D = A(16x4 f32) * B(4x16 f32) + C(16x16 f32)
  return __builtin_amdgcn_wmma_f32_16x16x4_f32(false, a, false, b, (short)0, c,
                                               false, false);
}

__device__ __forceinline__ void atomAddF(float* p, float v) {
  __hip_atomic_fetch_add(p, v, __ATOMIC_RELAXED, __HIP_MEMORY_SCOPE_AGENT);
}

__device__ __forceinline__ float leakyf(float x) {
  return x > 0.0f ? x : 0.01f * x;
}

// ---------------------------------------------------------------------------
// Per-layer edge kernel: one wave processes a tile of 16 edges.
//   h_src     = leaky(hidden_raw[src])            (leaky fused from prev layer)
//   embed_rel = h_src + rela[rel] + time[|rt|]                 [16 x 20]
//   transformed = embed_rel @ W{fut|now|past}.T (per-edge select on sign(rt))
//   score = sigmoid(relu([h_src, r_emb, q_emb] @ Wa1.T) @ Wa2.T)
//   atomicAdd hidden_out[dst] += score * transformed
// WMMA shape: M=16 edges, K chunks of 4, N tiles of 16 columns.
// B operands are pre-swizzled into LDS in the exact per-lane fragment layout
// [chunk][half][col(32, zero-padded)][2] so each B fetch is one ds_load_b64
// with no divergent guards around the WMMAs.
// ---------------------------------------------------------------------------
__global__ __launch_bounds__(256) void tgnn_layer_kernel(
    const int* __restrict__ src, const int* __restrict__ dst,
    const int* __restrict__ rel, const int* __restrict__ qrel,
    const int* __restrict__ rtm, const float* __restrict__ hin,
    float* __restrict__ hout, const float* __restrict__ rela,
    const float* __restrict__ temb, const float* __restrict__ wa1,
    const float* __restrict__ wa2, const float* __restrict__ wp,
    const float* __restrict__ wn, const float* __restrict__ wf, int E) {
  // B fragments: sWB[w][c][half][n][x] = W_w[n][4c+2*half+x] (n<20 else 0)
  __shared__ float sWB[3][5][2][32][2];   // 1920 floats
  __shared__ float sAB[15][2][32][2];     // 1920 floats: Wa1[n][4c+2*half+x]
  __shared__ float sA2[32];

  const int tid = threadIdx.x, bs = blockDim.x;
  {
    const float* Ws[3] = {wf, wn, wp};    // order: future, now, past
    for (int i = tid; i < 3 * 640; i += bs) {
      const int w = i / 640, rem = i % 640;
      const int c = rem / 128, r2 = rem % 128;
      const int h = r2 / 64, r3 = r2 % 64;
      const int n = r3 / 2, x = r3 % 2;
      const int k = 4 * c + 2 * h + x;
      sWB[w][c][h][n][x] = (n < D_EMB) ? Ws[w][n * D_EMB + k] : 0.0f;
    }
    for (int i = tid; i < 1920; i += bs) {
      const int c = i / 128, r2 = i % 128;
      const int h = r2 / 64, r3 = r2 % 64;
      const int n = r3 / 2, x = r3 % 2;
      const int k = 4 * c + 2 * h + x;
      sAB[c][h][n][x] = (n < 30) ? wa1[n * 60 + k] : 0.0f;
    }
    if (tid < 32) sA2[tid] = (tid < 30) ? wa2[tid] : 0.0f;
  }
  __syncthreads();

  const int lane = tid & 31;
  const int m = lane & 15;     // edge-in-tile for A, column for B/D
  const int half = lane >> 4;  // K-half (A/B), row-block (D)
  const long long tile = (long long)blockIdx.x * (bs >> 5) + (tid >> 5);
  const long long ebase = tile * 16;
  if (ebase >= E) return;

  const long long e = ebase + m;
  const bool val = e < (long long)E;
  const int s_ = val ? src[e] : 0;
  const int d_ = val ? dst[e] : 0;
  const int r_ = val ? rel[e] : 0;
  const int q_ = val ? qrel[e] : 0;
  const int t_ = val ? rtm[e] : 0;
  const int ta = t_ < 0 ? -t_ : t_;

  // A-fragments: lane holds A[m][4c + 2*half + {0,1}]  (ISA f32 16x4 layout)
  v2f hf[5], rf[5], tf[5], qf[5], ef[5];
#pragma unroll
  for (int c = 0; c < 5; ++c) {
    const int k = 4 * c + 2 * half;
    v2f zh = {}, zr = {}, zt = {}, zq = {};
    if (val) {
      zh = *(const v2f*)(hin + (long long)s_ * D_EMB + k);
      zr = *(const v2f*)(rela + (long long)r_ * D_EMB + k);
      zt = *(const v2f*)(temb + (long long)ta * D_EMB + k);
      zq = *(const v2f*)(rela + (long long)q_ * D_EMB + k);
    }
    zh.x = leakyf(zh.x);  // fused leaky-relu from previous layer
    zh.y = leakyf(zh.y);
    hf[c] = zh;
    rf[c] = zr;
    tf[c] = zt;
    qf[c] = zq;
    ef[c] = zh + zr + zt;
  }

  // Per-row (edge) metadata: D row j lives in lane-half `half`, edge j+8*half.
  int rt_row[8], dst_row[8], ev_row[8];
#pragma unroll
  for (int j = 0; j < 8; ++j) {
    const int srcLane = j + 8 * half;
    rt_row[j] = __shfl(t_, srcLane, 32);
    dst_row[j] = __shfl(d_, srcLane, 32);
    ev_row[j] = (ebase + srcLane) < (long long)E;
  }

  const int n0 = m;       // output-column tile 0 (dims 0..15)
  const int n1 = 16 + m;  // output-column tile 1 (dims 16..19, zero-padded)

  // Branch GEMM: acc = embed_rel @ W.T for one 16-wide N tile (unconditional
  // ds_load_b64 B fetches from the pre-padded fragment layout).
  auto gemmW = [&](int w, int ncol) -> v8f {
    v8f acc = {};
#pragma unroll
    for (int c = 0; c < 5; ++c) {
      const v2f b = *(const v2f*)&sWB[w][c][half][ncol][0];
      acc = wmma_f32_4(ef[c], b, acc);
    }
    return acc;
  };

  const v8f aF0 = gemmW(0, n0), aF1 = gemmW(0, n1);
  const v8f aN0 = gemmW(1, n0), aN1 = gemmW(1, n1);
  const v8f aP0 = gemmW(2, n0), aP1 = gemmW(2, n1);

  v8f sel0 = {}, sel1 = {};
#pragma unroll
  for (int j = 0; j < 8; ++j) {
    const int rtj = rt_row[j];
    sel0[j] = rtj > 0 ? aF0[j] : (rtj == 0 ? aN0[j] : aP0[j]);
    sel1[j] = rtj > 0 ? aF1[j] : (rtj == 0 ? aN1[j] : aP1[j]);
  }

  // Attention GEMM: [h_src | r_emb | q_emb] (K=60) @ Wa1.T (30 units).
  auto gemmA = [&](int ucol) -> v8f {
    v8f acc = {};
#pragma unroll
    for (int c = 0; c < 15; ++c) {
      const v2f a = (c < 5) ? hf[c] : (c < 10 ? rf[c - 5] : qf[c - 10]);
      const v2f b = *(const v2f*)&sAB[c][half][ucol][0];
      acc = wmma_f32_4(a, b, acc);
    }
    return acc;
  };
  const v8f at0 = gemmA(n0);  // units 0..15
  const v8f at1 = gemmA(n1);  // units 16..29 (30,31 zero-padded)

  // relu, dot with w_att2, 16-lane butterfly reduce per half, sigmoid.
  const float w2lo = sA2[m], w2hi = sA2[16 + m];
  float score[8];
#pragma unroll
  for (int j = 0; j < 8; ++j) {
    float p = fmaxf(at0[j], 0.0f) * w2lo + fmaxf(at1[j], 0.0f) * w2hi;
    p += __shfl_xor(p, 1, 32);
    p += __shfl_xor(p, 2, 32);
    p += __shfl_xor(p, 4, 32);
    p += __shfl_xor(p, 8, 32);
    score[j] = 1.0f / (1.0f + __expf(-p));
  }

  // Scatter: hidden_out[dst][n] += score * transformed  (hardware f32 atomics)
#pragma unroll
  for (int j = 0; j < 8; ++j) {
    if (!ev_row[j]) continue;
    const float sc = score[j];
    const long long base = (long long)dst_row[j] * D_EMB;
    atomAddF(&hout[base + m], sc * sel0[j]);
    if (m < 4) atomAddF(&hout[base + 16 + m], sc * sel1[j]);
  }
}

// ---------------------------------------------------------------------------
__global__ void zero4_kernel(float4* __restrict__ p, long long n4) {
  long long i = (long long)blockIdx.x * blockDim.x + threadIdx.x;
  const long long stride = (long long)gridDim.x * blockDim.x;
  const float4 z = make_float4(0.f, 0.f, 0.f, 0.f);
  for (; i < n4; i += stride) p[i] = z;
}

__global__ void classify_scatter_kernel(
    const float* __restrict__ hidden, const float* __restrict__ wcls,
    const float* __restrict__ bcls, const int* __restrict__ nbatch,
    const int* __restrict__ nent, const int* __restrict__ n_ent_ptr,
    int n_nodes, float* __restrict__ out) {
  const int n = blockIdx.x * blockDim.x + threadIdx.x;
  if (n >= n_nodes) return;
  const long long NE = (long long)(*n_ent_ptr);
  float acc = bcls[0];
  const float* h = hidden + (long long)n * D_EMB;
#pragma unroll
  for (int k = 0; k < D_EMB; ++k) acc += leakyf(h[k]) * wcls[k];
  out[(long long)nbatch[n] * NE + (long long)nent[n]] = acc;
}

// ---------------------------------------------------------------------------
extern "C" void kernel_launch(void* const* d_in, const int* in_sizes, int n_in,
                              void* d_out, int out_size, void* d_ws,
                              size_t ws_size, hipStream_t stream) {
  (void)n_in;
  (void)ws_size;
  const int* src = (const int*)d_in[0];
  const int* dst = (const int*)d_in[1];
  const int* rel = (const int*)d_in[2];
  const int* qrel = (const int*)d_in[3];
  const int* rtm = (const int*)d_in[4];
  const int* nbatch = (const int*)d_in[5];
  const int* nent = (const int*)d_in[6];
  const float* rela = (const float*)d_in[7];
  const float* temb = (const float*)d_in[8];
  const float* wa1 = (const float*)d_in[9];
  const float* wa2 = (const float*)d_in[10];
  const float* wp = (const float*)d_in[11];
  const float* wn = (const float*)d_in[12];
  const float* wf = (const float*)d_in[13];
  const float* wcls = (const float*)d_in[14];
  const float* bcls = (const float*)d_in[15];
  // d_in[16] = batch_size (device int scalar, unused), d_in[17] = n_ent
  const int* n_ent_ptr = (const int*)d_in[17];

  const int L = 3;
  const int E = in_sizes[0] / L;
  const int Nn = in_sizes[5];

  float* hA = (float*)d_ws;             // 80 MB ping (raw pre-leaky hidden)
  float* hB = hA + (size_t)Nn * D_EMB;  // 80 MB pong

  const long long hElems = (long long)Nn * D_EMB;       // divisible by 4
  const long long h4 = hElems / 4;
  const long long o4 = (long long)out_size / 4;         // 64e6+ div by 4
  const int zb = 256;
  int zgH = (int)((h4 + zb - 1) / zb);
  if (zgH > 1048576) zgH = 1048576;
  int zgO = (int)((o4 + zb - 1) / zb);
  if (zgO > 1048576) zgO = 1048576;

  zero4_kernel<<<zgH, zb, 0, stream>>>((float4*)hA, h4);  // layer-0 hidden = 0
  zero4_kernel<<<zgO, zb, 0, stream>>>((float4*)d_out, o4);

  const int wavesPerBlock = 256 / 32;
  const long long tiles = ((long long)E + 15) / 16;
  const int blocks = (int)((tiles + wavesPerBlock - 1) / wavesPerBlock);

  for (int l = 0; l < L; ++l) {
    zero4_kernel<<<zgH, zb, 0, stream>>>((float4*)hB, h4);
    tgnn_layer_kernel<<<blocks, 256, 0, stream>>>(
        src + (size_t)l * E, dst + (size_t)l * E, rel + (size_t)l * E,
        qrel + (size_t)l * E, rtm + (size_t)l * E, hA, hB, rela, temb, wa1,
        wa2, wp, wn, wf, E);
    float* t = hA;
    hA = hB;
    hB = t;
  }

  classify_scatter_kernel<<<(Nn + 255) / 256, 256, 0, stream>>>(
      hA, wcls, bcls, nbatch, nent, n_ent_ptr, Nn, (float*)d_out);
}